// NGramRepeatBlock_335007449599
// MI455X (gfx1250) — compile-verified
//
#include <hip/hip_runtime.h>
#include <hip/hip_bf16.h>

// NGramRepeatBlock for MI455X (gfx1250).
// Bandwidth-bound: ~206 MB streamed @ 23.3 TB/s => ~9us floor. No matrix math.
// Pass 1: non-temporal b128 streaming copy of lprobs -> out (exact grid,
//         unguarded fast path, 128 B per thread for deep MLP under wave32).
// Pass 2: tiny per-row bigram scan that scatters -inf for banned tokens.

typedef __attribute__((ext_vector_type(4))) float float4v;

#define COPY_THREADS 256
#define COPY_UNROLL  8   // 8 x float4 = 128 bytes per thread

// ---------------------------------------------------------------------------
// Kernel 1: streaming copy. 32-bit indexing (counts < 2^31 guaranteed: the
// harness passes out_size as int). Fast path has zero per-access guards.
// ---------------------------------------------------------------------------
__global__ __launch_bounds__(COPY_THREADS) void ngram_copy_kernel(
    const float4v* __restrict__ src4,
    float4v* __restrict__ dst4,
    const float* __restrict__ src_s,
    float* __restrict__ dst_s,
    unsigned n4,       // number of float4 elements
    unsigned n_tail)   // scalar remainder elements after n4*4
{
    const unsigned base = blockIdx.x * (COPY_THREADS * COPY_UNROLL) + threadIdx.x;

    if (base + COPY_THREADS * (COPY_UNROLL - 1) < n4) {
        // Full tile: 8 independent NT b128 loads, then 8 NT b128 stores.
        float4v v[COPY_UNROLL];
#pragma unroll
        for (int u = 0; u < COPY_UNROLL; ++u)
            v[u] = __builtin_nontemporal_load(src4 + base + COPY_THREADS * u);
#pragma unroll
        for (int u = 0; u < COPY_UNROLL; ++u)
            __builtin_nontemporal_store(v[u], dst4 + base + COPY_THREADS * u);
    } else {
        // Edge tile (last block only): per-access guard.
#pragma unroll
        for (int u = 0; u < COPY_UNROLL; ++u) {
            const unsigned i = base + COPY_THREADS * u;
            if (i < n4)
                __builtin_nontemporal_store(__builtin_nontemporal_load(src4 + i),
                                            dst4 + i);
        }
    }

    // Scalar tail (< 4 elements); a few threads of block 0.
    if (blockIdx.x == 0 && threadIdx.x < n_tail) {
        const unsigned i = n4 * 4u + threadIdx.x;
        dst_s[i] = src_s[i];
    }
}

// ---------------------------------------------------------------------------
// Kernel 2: per-row bigram scan + sparse -inf scatter.
// Reads bsz/beam_size/step from device scalar buffers; derives L and V from
// the flat element counts, so no host-side knowledge of B is needed.
// ---------------------------------------------------------------------------
__global__ __launch_bounds__(256) void ngram_ban_kernel(
    const int* __restrict__ tokens,
    float* __restrict__ out,
    const int* __restrict__ p_bsz,
    const int* __restrict__ p_beam,
    const int* __restrict__ p_step,
    unsigned tokens_total,
    unsigned lprobs_total)
{
    const int NGRAM = 3;
    const int B    = p_bsz[0] * p_beam[0];
    const int step = p_step[0];
    if (B <= 0) return;

    const unsigned L = tokens_total / (unsigned)B;
    const unsigned V = lprobs_total / (unsigned)B;

    const int check = step + 2 - NGRAM;   // number of candidate window starts
    if (check <= 0 || L < (unsigned)NGRAM) return;

    const float neg_inf = -__builtin_inff();

    for (unsigned row = blockIdx.x; row < (unsigned)B; row += gridDim.x) {
        const int* __restrict__ t = tokens + (size_t)row * L;
        // Prefix = last (n-1)=2 tokens of the row.
        const int p0 = t[L - 2];
        const int p1 = t[L - 1];
        float* __restrict__ orow = out + (size_t)row * V;

        for (int i = threadIdx.x; i < check; i += blockDim.x) {
            // Keep the scan streaming ahead (gfx1250 global_prefetch_b8).
            __builtin_prefetch(t + i + 4 * (int)blockDim.x, 0, 1);

            if (t[i] == p0 && t[i + 1] == p1) {
                const int id = t[i + 2];
                if (id >= 0 && (unsigned)id < V) {
                    orow[id] = neg_inf;   // idempotent; races between matches OK
                }
            }
        }
    }
}

// ---------------------------------------------------------------------------
// Launch
// ---------------------------------------------------------------------------
extern "C" void kernel_launch(void* const* d_in, const int* in_sizes, int n_in,
                              void* d_out, int out_size, void* d_ws, size_t ws_size,
                              hipStream_t stream) {
    (void)n_in; (void)d_ws; (void)ws_size; (void)out_size;

    const int*   tokens = (const int*)d_in[0];     // (B, L) ints
    const float* lprobs = (const float*)d_in[1];   // (B, V) f32
    const int*   p_bsz  = (const int*)d_in[2];     // scalar (device)
    const int*   p_beam = (const int*)d_in[3];     // scalar (device)
    const int*   p_step = (const int*)d_in[4];     // scalar (device)
    float*       out    = (float*)d_out;           // (B, V) f32

    const unsigned tokens_total = (unsigned)in_sizes[0];
    const unsigned lprobs_total = (unsigned)in_sizes[1];

    // ---- Pass 1: streaming NT copy of lprobs -> out (exact grid) ----
    const unsigned n4     = lprobs_total / 4u;
    const unsigned n_tail = lprobs_total - n4 * 4u;
    const unsigned per_block = COPY_THREADS * COPY_UNROLL;        // 2048 float4
    unsigned blocks = (n4 + per_block - 1) / per_block;
    if (blocks == 0) blocks = 1;

    ngram_copy_kernel<<<blocks, COPY_THREADS, 0, stream>>>(
        (const float4v*)lprobs, (float4v*)out,
        lprobs, out, n4, n_tail);

    // ---- Pass 2: bigram-match ban scatter (after copy via stream order) ----
    ngram_ban_kernel<<<1024, 256, 0, stream>>>(
        tokens, out, p_bsz, p_beam, p_step, tokens_total, lprobs_total);
}